// SlotAttention_13168369729822
// MI455X (gfx1250) — compile-verified
//
#include <hip/hip_runtime.h>
#include <math.h>

typedef __attribute__((ext_vector_type(2))) float v2f;
typedef __attribute__((ext_vector_type(8))) float v8f;

#define EMB_D   768
#define N_SLOTS 16
#define KEY_D   64
#define N_ITERS 3
#define N_TOK   196
#define BS      128
#define PDIM    30720   /* 5*8*768 */
#define LNS     772     /* padded LDS row stride (dwords) for feature tiles */

static __device__ inline v8f wmma_f32_16x16x4(v2f a, v2f b, v8f c) {
  // D = A(16x4,f32) * B(4x16,f32) + C(16x16,f32)
  return __builtin_amdgcn_wmma_f32_16x16x4_f32(false, a, false, b, (short)0, c,
                                               false, false);
}

// ---------------------------------------------------------------------------
// Kernel 1: fused LayerNorm(features) + K = f@Wk, V = f@Wv
// grid.x = 25088/16 row-blocks, 256 threads (8 waves)
// ---------------------------------------------------------------------------
__global__ __launch_bounds__(256) void k1_ln_kv(
    const float* __restrict__ features, const float* __restrict__ lnw,
    const float* __restrict__ lnb, const float* __restrict__ Wk,
    const float* __restrict__ Wv, float* __restrict__ k_out,
    float* __restrict__ v_out) {
  __shared__ float fln[16 * LNS];
  const int tid = threadIdx.x;
  const int lane = tid & 31;
  const int wave = tid >> 5;
  const int rowblk = blockIdx.x;

  // ---- LayerNorm: each wave normalizes 2 rows of 768 ----
  for (int r = wave * 2; r < wave * 2 + 2; ++r) {
    const float* src = features + (size_t)(rowblk * 16 + r) * EMB_D;
    float s = 0.f, ss = 0.f;
    float vals[EMB_D / 32];
#pragma unroll
    for (int j = 0; j < EMB_D / 32; ++j) {
      float x = src[lane + j * 32];
      vals[j] = x; s += x; ss += x * x;
    }
#pragma unroll
    for (int m = 16; m >= 1; m >>= 1) {
      s  += __shfl_xor(s, m, 32);
      ss += __shfl_xor(ss, m, 32);
    }
    float mean = s * (1.f / EMB_D);
    float var  = ss * (1.f / EMB_D) - mean * mean;
    float rstd = rsqrtf(var + 1e-5f);
#pragma unroll
    for (int j = 0; j < EMB_D / 32; ++j) {
      int c = lane + j * 32;
      fln[r * LNS + c] = (vals[j] - mean) * rstd * lnw[c] + lnb[c];
    }
  }
  __syncthreads();

  // ---- GEMM: wave -> (K or V) x 16-col tile, K-dim = 768 in steps of 4 ----
  const float* W = (wave < 4) ? Wk : Wv;
  float* out     = (wave < 4) ? k_out : v_out;
  const int ct  = wave & 3;
  const int m   = lane & 15;
  const int off = (lane < 16) ? 0 : 2;
  v8f acc = {0.f, 0.f, 0.f, 0.f, 0.f, 0.f, 0.f, 0.f};
  for (int k0 = 0; k0 < EMB_D; k0 += 4) {
    v2f a, bf;
    a.x  = fln[m * LNS + k0 + off];
    a.y  = fln[m * LNS + k0 + off + 1];
    bf.x = W[(k0 + off) * KEY_D + ct * 16 + m];
    bf.y = W[(k0 + off + 1) * KEY_D + ct * 16 + m];
    acc  = wmma_f32_16x16x4(a, bf, acc);
  }
#pragma unroll
  for (int i = 0; i < 8; ++i) {
    int mr = (lane < 16) ? i : (i + 8);
    out[(size_t)(rowblk * 16 + mr) * KEY_D + ct * 16 + m] = acc[i];
  }
}

// ---------------------------------------------------------------------------
// Kernel 2: full 3-iteration slot-attention loop, one workgroup per batch
// ---------------------------------------------------------------------------
__global__ __launch_bounds__(256) void k2_slots(
    const float* __restrict__ k_g, const float* __restrict__ v_g,
    const float* __restrict__ noise, const float* __restrict__ mu,
    const float* __restrict__ sigma, const float* __restrict__ ln_slot_w,
    const float* __restrict__ ln_slot_b, const float* __restrict__ ln_out_w,
    const float* __restrict__ ln_out_b, const float* __restrict__ Wq,
    const float* __restrict__ W_ih, const float* __restrict__ W_hh,
    const float* __restrict__ b_ih, const float* __restrict__ b_hh,
    const float* __restrict__ W1, const float* __restrict__ b1,
    const float* __restrict__ W2, const float* __restrict__ b2,
    float* __restrict__ slots_out) {
  __shared__ float sl[N_SLOTS * KEY_D];     // current slots
  __shared__ float sp[N_SLOTS * KEY_D];     // slots_prev
  __shared__ float lnbuf[N_SLOTS * KEY_D];  // LN scratch (slots / out)
  __shared__ float qT[KEY_D * N_SLOTS];     // q transposed [d][slot]
  __shared__ float attn[208 * N_SLOTS];     // padded to 13 tiles of 16 rows
  __shared__ float upd[N_SLOTS * KEY_D];
  __shared__ float gi[N_SLOTS * 192];
  __shared__ float gh[N_SLOTS * 192];
  __shared__ float m1[N_SLOTS * KEY_D];
  __shared__ float colinv[N_SLOTS];

  const int tid = threadIdx.x;
  const int lane = tid & 31, wave = tid >> 5;
  const int b = blockIdx.x;
  const float* kb = k_g + (size_t)b * N_TOK * KEY_D;
  const float* vb = v_g + (size_t)b * N_TOK * KEY_D;

  for (int i = tid; i < N_SLOTS * KEY_D; i += 256) {
    int d = i & 63;
    sl[i] = noise[(size_t)b * N_SLOTS * KEY_D + i] * sigma[d] + mu[d];
  }
  __syncthreads();

  const float scale = 0.125f;  // KEY_D^-0.5

  for (int it = 0; it < N_ITERS; ++it) {
    for (int i = tid; i < 1024; i += 256) sp[i] = sl[i];
    __syncthreads();

    // LN(slots), one thread per slot row (tiny)
    if (tid < N_SLOTS) {
      float s = 0.f, ss = 0.f;
      for (int d = 0; d < KEY_D; ++d) { float x = sl[tid*64+d]; s += x; ss += x*x; }
      float mean = s * (1.f/64.f), var = ss * (1.f/64.f) - mean*mean;
      float rstd = rsqrtf(var + 1e-5f);
      for (int d = 0; d < KEY_D; ++d)
        lnbuf[tid*64+d] = (sl[tid*64+d]-mean)*rstd*ln_slot_w[d] + ln_slot_b[d];
    }
    __syncthreads();

    // q^T[d_out][slot] = (LN(slots) @ Wq)^T
    for (int i = tid; i < 1024; i += 256) {
      int s_ = i >> 6, o = i & 63;
      float a = 0.f;
      for (int d = 0; d < 64; ++d) a += lnbuf[s_*64+d] * Wq[d*64+o];
      qT[o*16 + s_] = a;
    }
    __syncthreads();

    // attn logits [196(pad 208) x 16] = k @ q^T * scale, via WMMA
    {
      const int m_  = lane & 15;
      const int off = (lane < 16) ? 0 : 2;
      for (int t = wave; t < 13; t += 8) {
        v8f acc = {0,0,0,0,0,0,0,0};
        const int tok = t * 16 + m_;
        for (int k0 = 0; k0 < KEY_D; k0 += 4) {
          v2f a, bf;
          a.x  = (tok < N_TOK) ? kb[tok*64 + k0 + off]     : 0.f;
          a.y  = (tok < N_TOK) ? kb[tok*64 + k0 + off + 1] : 0.f;
          bf.x = qT[(k0+off)*16 + m_];
          bf.y = qT[(k0+off+1)*16 + m_];
          acc  = wmma_f32_16x16x4(a, bf, acc);
        }
#pragma unroll
        for (int i = 0; i < 8; ++i) {
          int tr = t*16 + ((lane < 16) ? i : i + 8);
          attn[tr*16 + m_] = acc[i] * scale;
        }
      }
    }
    __syncthreads();

    // softmax over slots, + eps
    if (tid < N_TOK) {
      float mx = -1e30f;
      for (int s_ = 0; s_ < 16; ++s_) mx = fmaxf(mx, attn[tid*16+s_]);
      float e[16], sum = 0.f;
      for (int s_ = 0; s_ < 16; ++s_) { e[s_] = __expf(attn[tid*16+s_]-mx); sum += e[s_]; }
      float inv = 1.f / sum;
      for (int s_ = 0; s_ < 16; ++s_) attn[tid*16+s_] = e[s_]*inv + 1e-8f;
    }
    __syncthreads();

    // renormalize over tokens
    if (tid < N_SLOTS) {
      float s = 0.f;
      for (int n = 0; n < N_TOK; ++n) s += attn[n*16+tid];
      colinv[tid] = 1.f / s;
    }
    __syncthreads();
    for (int i = tid; i < N_TOK*16; i += 256) attn[i] *= colinv[i & 15];
    __syncthreads();

    // updates[16x64] = attn^T @ v, waves 0..3 (one 16-col tile each), K=196
    if (wave < 4) {
      const int m_  = lane & 15;
      const int off = (lane < 16) ? 0 : 2;
      v8f acc = {0,0,0,0,0,0,0,0};
      for (int k0 = 0; k0 < N_TOK; k0 += 4) {
        v2f a, bf;
        a.x  = attn[(k0+off)*16 + m_];
        a.y  = attn[(k0+off+1)*16 + m_];
        bf.x = vb[(k0+off)*64 + wave*16 + m_];
        bf.y = vb[(k0+off+1)*64 + wave*16 + m_];
        acc  = wmma_f32_16x16x4(a, bf, acc);
      }
#pragma unroll
      for (int i = 0; i < 8; ++i) {
        int sr = (lane < 16) ? i : i + 8;
        upd[sr*64 + wave*16 + m_] = acc[i];
      }
    }
    __syncthreads();

    // GRU gate pre-activations (PyTorch r,z,n order)
    for (int i = tid; i < N_SLOTS*192; i += 256) {
      int s_ = i / 192, j = i % 192;
      float ai = b_ih[j], ah = b_hh[j];
      for (int d = 0; d < 64; ++d) {
        ai += upd[s_*64+d] * W_ih[j*64+d];
        ah += sp[s_*64+d] * W_hh[j*64+d];
      }
      gi[i] = ai; gh[i] = ah;
    }
    __syncthreads();
    for (int i = tid; i < 1024; i += 256) {
      int s_ = i >> 6, d = i & 63;
      float r = 1.f/(1.f + __expf(-(gi[s_*192+d]      + gh[s_*192+d])));
      float z = 1.f/(1.f + __expf(-(gi[s_*192+64+d]   + gh[s_*192+64+d])));
      float n = tanhf(gi[s_*192+128+d] + r * gh[s_*192+128+d]);
      sl[i] = (1.f - z) * n + z * sp[i];
    }
    __syncthreads();

    // residual MLP: slots += relu(LN(slots)@W1^T + b1)@W2^T + b2
    if (tid < N_SLOTS) {
      float s = 0.f, ss = 0.f;
      for (int d = 0; d < 64; ++d) { float x = sl[tid*64+d]; s += x; ss += x*x; }
      float mean = s * (1.f/64.f), var = ss * (1.f/64.f) - mean*mean;
      float rstd = rsqrtf(var + 1e-5f);
      for (int d = 0; d < 64; ++d)
        lnbuf[tid*64+d] = (sl[tid*64+d]-mean)*rstd*ln_out_w[d] + ln_out_b[d];
    }
    __syncthreads();
    for (int i = tid; i < 1024; i += 256) {
      int s_ = i >> 6, j = i & 63;
      float a = b1[j];
      for (int d = 0; d < 64; ++d) a += lnbuf[s_*64+d] * W1[j*64+d];
      m1[i] = fmaxf(a, 0.f);
    }
    __syncthreads();
    for (int i = tid; i < 1024; i += 256) {
      int s_ = i >> 6, o = i & 63;
      float a = b2[o];
      for (int j = 0; j < 64; ++j) a += m1[s_*64+j] * W2[o*64+j];
      sl[i] += a;
    }
    __syncthreads();
  }

  for (int i = tid; i < 1024; i += 256)
    slots_out[(size_t)b*1024 + i] = sl[i];
}

// ---------------------------------------------------------------------------
// Kernel 3: prompts[2048 x 30720] = slots @ s2p_flat  (streaming WMMA GEMM)
// grid = (128 batches, 240), 8 waves -> 8 N-tiles of 16 cols each
// ---------------------------------------------------------------------------
__global__ __launch_bounds__(256) void k3_prompts(
    const float* __restrict__ slots, const float* __restrict__ s2p,
    float* __restrict__ out) {
  const int tid  = threadIdx.x;
  const int lane = tid & 31, wave = tid >> 5;
  const int b = blockIdx.x;
  const int ntile = blockIdx.y * 8 + wave;
  const int m_  = lane & 15;
  const int off = (lane < 16) ? 0 : 2;
  const int col = ntile * 16 + m_;
  const float* slb = slots + (size_t)b * N_SLOTS * KEY_D;

  v2f a[16];
#pragma unroll
  for (int kk = 0; kk < 16; ++kk) {
    a[kk].x = slb[m_*64 + kk*4 + off];
    a[kk].y = slb[m_*64 + kk*4 + off + 1];
  }
  v8f acc = {0,0,0,0,0,0,0,0};
#pragma unroll
  for (int kk = 0; kk < 16; ++kk) {
    v2f bf;
    bf.x = s2p[(size_t)(kk*4 + off) * PDIM + col];
    bf.y = s2p[(size_t)(kk*4 + off + 1) * PDIM + col];
    acc  = wmma_f32_16x16x4(a[kk], bf, acc);
  }
#pragma unroll
  for (int i = 0; i < 8; ++i) {
    int mr = (lane < 16) ? i : i + 8;
    out[(size_t)(b * 16 + mr) * PDIM + col] = acc[i];
  }
}

// ---------------------------------------------------------------------------
extern "C" void kernel_launch(void* const* d_in, const int* in_sizes, int n_in,
                              void* d_out, int out_size, void* d_ws,
                              size_t ws_size, hipStream_t stream) {
  const float* features  = (const float*)d_in[0];
  const float* noise     = (const float*)d_in[1];
  const float* ln_in_w   = (const float*)d_in[2];
  const float* ln_in_b   = (const float*)d_in[3];
  const float* ln_slot_w = (const float*)d_in[4];
  const float* ln_slot_b = (const float*)d_in[5];
  const float* ln_out_w  = (const float*)d_in[6];
  const float* ln_out_b  = (const float*)d_in[7];
  const float* mu        = (const float*)d_in[8];
  const float* sigma     = (const float*)d_in[9];
  const float* Wk        = (const float*)d_in[10];
  const float* Wq        = (const float*)d_in[11];
  const float* Wv        = (const float*)d_in[12];
  const float* W_ih      = (const float*)d_in[13];
  const float* W_hh      = (const float*)d_in[14];
  const float* b_ih      = (const float*)d_in[15];
  const float* b_hh      = (const float*)d_in[16];
  const float* W1        = (const float*)d_in[17];
  const float* b1        = (const float*)d_in[18];
  const float* W2        = (const float*)d_in[19];
  const float* b2        = (const float*)d_in[20];
  const float* s2p       = (const float*)d_in[21];

  float* k_ws     = (float*)d_ws;
  float* v_ws     = k_ws + (size_t)BS * N_TOK * KEY_D;
  float* slots_ws = v_ws + (size_t)BS * N_TOK * KEY_D;

  k1_ln_kv<<<dim3(BS * N_TOK / 16), 256, 0, stream>>>(
      features, ln_in_w, ln_in_b, Wk, Wv, k_ws, v_ws);
  k2_slots<<<dim3(BS), 256, 0, stream>>>(
      k_ws, v_ws, noise, mu, sigma, ln_slot_w, ln_slot_b, ln_out_w, ln_out_b,
      Wq, W_ih, W_hh, b_ih, b_hh, W1, b1, W2, b2, slots_ws);
  k3_prompts<<<dim3(BS, PDIM / 128), 256, 0, stream>>>(
      slots_ws, s2p, (float*)d_out);
}